// GCNGC_46651934769782
// MI455X (gfx1250) — compile-verified
//
#include <hip/hip_runtime.h>

typedef float v2f __attribute__((ext_vector_type(2)));
typedef float v8f __attribute__((ext_vector_type(8)));

#define D_H 128
#define XROWS 64          // rows per block = 4 waves * 16
#define SX_STRIDE 132     // floats; 132 % 64 = 4 -> conflict-free A b64 reads
#define SW_STRIDE 288     // floats per k-pair row; 288 % 64 = 32 -> halves disjoint

// ---------------------------------------------------------------------------
// Dense GEMM, K=128, Ncols=128, fp32 WMMA 16x16x4, LDS-staged operands.
//   A frag : lane m=lane&15, K-half by lane>>4  -> ds_load_b64 from sX
//   B frag : sW packed as [p][n][k&1] (p=k/2)   -> ds_load_b64 per tile
//   C/D    : VGPR v -> row v + 8*(lane>>4), col lane&15
// ---------------------------------------------------------------------------
__global__ __launch_bounds__(128) void gemm128_lds_wmma(
    const float* __restrict__ X, const float* __restrict__ W,
    const float* __restrict__ bias, float* __restrict__ Y, int nrows,
    int relu_in) {
  __shared__ float sW[64 * SW_STRIDE];      // 73.7 KB packed weights
  __shared__ float sX[XROWS * SX_STRIDE];   // 33.8 KB activation tile

  const int tid = threadIdx.x;
  const int wave = tid >> 5;
  const int lane = tid & 31;
  const int row0 = blockIdx.x * XROWS + wave * 16;

  // ---- stage W (coalesced b128 reads, packed ds stores) ----
  for (int i = tid * 4; i < D_H * D_H; i += 128 * 4) {
    const float4 w4 = *reinterpret_cast<const float4*>(W + i);
    const int k = i >> 7;        // W row
    const int n = i & 127;       // W col (multiple of 4)
    float* dst = sW + (k >> 1) * SW_STRIDE + (k & 1);
    dst[2 * (n + 0)] = w4.x;
    dst[2 * (n + 1)] = w4.y;
    dst[2 * (n + 2)] = w4.z;
    dst[2 * (n + 3)] = w4.w;
  }
  // ---- stage X tile rows (coalesced b128 in, b128 out), fused relu ----
  for (int i = tid; i < XROWS * 32; i += 128) {
    const int r = i >> 5;        // local row
    const int c4 = i & 31;       // float4 column
    int gr = blockIdx.x * XROWS + r;
    if (gr >= nrows) gr = nrows - 1;   // clamp: loads stay in-bounds
    float4 v = *reinterpret_cast<const float4*>(X + (size_t)gr * D_H + c4 * 4);
    if (relu_in) {
      v.x = fmaxf(v.x, 0.0f);
      v.y = fmaxf(v.y, 0.0f);
      v.z = fmaxf(v.z, 0.0f);
      v.w = fmaxf(v.w, 0.0f);
    }
    *reinterpret_cast<float4*>(sX + r * SX_STRIDE + c4 * 4) = v;
  }
  __syncthreads();

  if (row0 >= nrows) return;  // wave-uniform: EXEC all-1s around WMMA

  const int half = lane >> 4;
  const int l15 = lane & 15;
  const float* __restrict__ aRow = sX + (wave * 16 + l15) * SX_STRIDE;

  v8f acc[8];
#pragma unroll
  for (int t = 0; t < 8; ++t)
#pragma unroll
    for (int v = 0; v < 8; ++v) acc[t][v] = 0.0f;

#pragma unroll 4
  for (int k0 = 0; k0 < D_H; k0 += 4) {
    const int ka = k0 + 2 * half;
    const v2f a = *reinterpret_cast<const v2f*>(aRow + ka);         // ds_load_b64
    const float* __restrict__ bRow = sW + (ka >> 1) * SW_STRIDE;
#pragma unroll
    for (int t = 0; t < 8; ++t) {
      const v2f b =
          *reinterpret_cast<const v2f*>(bRow + 2 * (t * 16 + l15)); // ds_load_b64
      acc[t] = __builtin_amdgcn_wmma_f32_16x16x4_f32(
          false, a, false, b, (short)0, acc[t], false, false);
    }
  }

#pragma unroll
  for (int t = 0; t < 8; ++t) {
    const int col = t * 16 + l15;
    const float bv = bias[col];
#pragma unroll
    for (int v = 0; v < 8; ++v) {
      const int r = row0 + v + 8 * half;
      if (r < nrows) Y[(size_t)r * D_H + col] = acc[t][v] + bv;
    }
  }
}

// ---------------------------------------------------------------------------
// Head GEMM: pooled[128,128] @ Wout[128,10] + bout. Tiny; direct-global WMMA.
// ---------------------------------------------------------------------------
__global__ __launch_bounds__(128) void gemm_head_wmma(
    const float* __restrict__ X, const float* __restrict__ W,
    const float* __restrict__ bias, float* __restrict__ Y, int nrows,
    int ncols) {
  const int wave = threadIdx.x >> 5;
  const int lane = threadIdx.x & 31;
  const int row0 = (blockIdx.x * 4 + wave) * 16;
  if (row0 >= nrows) return;

  const int half = lane >> 4;
  const int l15 = lane & 15;
  int arow = row0 + l15;
  if (arow >= nrows) arow = nrows - 1;
  const float* __restrict__ aptr = X + (size_t)arow * D_H;

  v8f acc;
#pragma unroll
  for (int v = 0; v < 8; ++v) acc[v] = 0.0f;

  for (int k0 = 0; k0 < D_H; k0 += 4) {
    const int ka = k0 + 2 * half;
    v2f a;
    a[0] = aptr[ka];
    a[1] = aptr[ka + 1];
    v2f b;
    b[0] = (l15 < ncols) ? W[(size_t)ka * ncols + l15] : 0.0f;
    b[1] = (l15 < ncols) ? W[(size_t)(ka + 1) * ncols + l15] : 0.0f;
    acc = __builtin_amdgcn_wmma_f32_16x16x4_f32(false, a, false, b, (short)0,
                                                acc, false, false);
  }

  if (l15 < ncols) {
    const float bv = bias[l15];
#pragma unroll
    for (int v = 0; v < 8; ++v) {
      const int r = row0 + v + 8 * half;
      if (r < nrows) Y[(size_t)r * ncols + l15] = acc[v] + bv;
    }
  }
}

// ---------------------------------------------------------------------------
// COO SpMM scatter: acc[row[e], :] += vals[e] * src[col[e], :]
// One wave per edge; b128 gather of source row, f32 atomics to L2.
// ---------------------------------------------------------------------------
__global__ __launch_bounds__(256) void spmm_scatter(
    const int* __restrict__ row, const int* __restrict__ col,
    const float* __restrict__ vals, const float* __restrict__ src,
    float* __restrict__ acc, int E) {
  const int e = blockIdx.x * 8 + (threadIdx.x >> 5);
  if (e >= E) return;
  const int lane = threadIdx.x & 31;
  const int r = row[e];
  const int c = col[e];
  const float v = vals[e];
  const float4 s = reinterpret_cast<const float4*>(src + (size_t)c * D_H)[lane];
  float* dst = acc + (size_t)r * D_H + lane * 4;
  atomicAdd(dst + 0, v * s.x);
  atomicAdd(dst + 1, v * s.y);
  atomicAdd(dst + 2, v * s.z);
  atomicAdd(dst + 3, v * s.w);
}

// ---------------------------------------------------------------------------
// Pooling: pooled[batch_index[n], :] += relu(H[n, :]) ; one wave per node.
// ---------------------------------------------------------------------------
__global__ __launch_bounds__(256) void pool_scatter(
    const float* __restrict__ H, const int* __restrict__ batch_index,
    float* __restrict__ pooled, int n) {
  const int node = blockIdx.x * 8 + (threadIdx.x >> 5);
  if (node >= n) return;
  const int lane = threadIdx.x & 31;
  const int g = batch_index[node];
  const float4 s = reinterpret_cast<const float4*>(H + (size_t)node * D_H)[lane];
  float* dst = pooled + (size_t)g * D_H + lane * 4;
  atomicAdd(dst + 0, fmaxf(s.x, 0.0f));
  atomicAdd(dst + 1, fmaxf(s.y, 0.0f));
  atomicAdd(dst + 2, fmaxf(s.z, 0.0f));
  atomicAdd(dst + 3, fmaxf(s.w, 0.0f));
}

// Vectorized zero fill (counts here are multiples of 4 floats).
__global__ __launch_bounds__(256) void zero_f32v4(float4* __restrict__ p,
                                                  long n4) {
  const long i = (long)blockIdx.x * 256 + threadIdx.x;
  if (i < n4) p[i] = make_float4(0.f, 0.f, 0.f, 0.f);
}

// ---------------------------------------------------------------------------
// Orchestration. Inputs (setup_inputs order):
//  0 x[N,128] f32 | 1 adj_row[E] i32 | 2 adj_col[E] i32 | 3 adj_vals[E] f32
//  4 batch_index[N] i32 | 5 W1 | 6 b1 | 7 W2 | 8 b2 | 9 Wout[128,10] | 10 bout
// Output: [B=128, 10] f32.
// ---------------------------------------------------------------------------
extern "C" void kernel_launch(void* const* d_in, const int* in_sizes, int n_in,
                              void* d_out, int out_size, void* d_ws,
                              size_t ws_size, hipStream_t stream) {
  const float* x = (const float*)d_in[0];
  const int* adj_row = (const int*)d_in[1];
  const int* adj_col = (const int*)d_in[2];
  const float* adj_vals = (const float*)d_in[3];
  const int* batch_index = (const int*)d_in[4];
  const float* W1 = (const float*)d_in[5];
  const float* b1 = (const float*)d_in[6];
  const float* W2 = (const float*)d_in[7];
  const float* b2 = (const float*)d_in[8];
  const float* Wout = (const float*)d_in[9];
  const float* bout = (const float*)d_in[10];
  float* out = (float*)d_out;

  const int N = in_sizes[0] / D_H;  // 50000
  const int E = in_sizes[1];        // 800000
  const int D_OUT = in_sizes[10];   // 10
  const int B = out_size / D_OUT;   // 128

  // Workspace layout: lin | acc | pooled (acc and pooled contiguous).
  float* lin = (float*)d_ws;
  float* acc = lin + (size_t)N * D_H;
  float* pooled = acc + (size_t)N * D_H;

  const int gemm_blocks = (N + XROWS - 1) / XROWS;
  const int spmm_blocks = (E + 7) / 8;
  const int pool_blocks = (N + 7) / 8;
  const long nh = (long)N * D_H;

  // Layer 1: lin = x @ W1 + b1
  gemm128_lds_wmma<<<gemm_blocks, 128, 0, stream>>>(x, W1, b1, lin, N, 0);
  // zero acc + pooled (contiguous region)
  {
    const long n4 = (nh + (long)B * D_H) / 4;
    zero_f32v4<<<(int)((n4 + 255) / 256), 256, 0, stream>>>((float4*)acc, n4);
  }
  // acc = adj @ lin (scatter-add)
  spmm_scatter<<<spmm_blocks, 256, 0, stream>>>(adj_row, adj_col, adj_vals,
                                                lin, acc, E);
  // Layer 2: lin = relu(acc) @ W2 + b2
  gemm128_lds_wmma<<<gemm_blocks, 128, 0, stream>>>(acc, W2, b2, lin, N, 1);
  // re-zero acc
  zero_f32v4<<<(int)((nh / 4 + 255) / 256), 256, 0, stream>>>((float4*)acc,
                                                              nh / 4);
  // acc = adj @ lin
  spmm_scatter<<<spmm_blocks, 256, 0, stream>>>(adj_row, adj_col, adj_vals,
                                                lin, acc, E);
  // pooled[g] += relu(acc[node])
  pool_scatter<<<pool_blocks, 256, 0, stream>>>(acc, batch_index, pooled, N);
  // out = pooled @ Wout + bout   (M=128, N=10, K=128)
  gemm_head_wmma<<<(B + 63) / 64, 128, 0, stream>>>(pooled, Wout, bout, out, B,
                                                    D_OUT);
}